// SampleAndGroup_7842610283202
// MI455X (gfx1250) — compile-verified
//
#include <hip/hip_runtime.h>
#include <hip/hip_bf16.h>

#define BATCH    4
#define NPTS     8192
#define NPOINT   2048
#define NSAMPLE  32
#define NFEAT    16
#define RADIUS2  1.0f

typedef float v2f __attribute__((ext_vector_type(2)));
typedef float v8f __attribute__((ext_vector_type(8)));

// ---------------------------------------------------------------------------
// 1) Farthest point sampling: one 1024-thread block per batch.
//    Distances live in registers (8 points/thread); per-iteration reduction is
//    wave32 shfl butterflies + 32-entry LDS hop. fps semantics match jax scan:
//    idxs[0]=0, idxs[i]=argmax after update with idxs[i-1].
// ---------------------------------------------------------------------------
__global__ __launch_bounds__(1024) void fps_kernel(const float* __restrict__ in,
                                                   int* __restrict__ fpsIdx) {
    const int b   = blockIdx.x;
    const int tid = threadIdx.x;
    const float* base = in + (size_t)b * NPTS * NFEAT;

    float px[8], py[8], pz[8], dist[8];
#pragma unroll
    for (int k = 0; k < 8; ++k) {
        int p = tid + k * 1024;
        px[k] = base[p * NFEAT + 0];
        py[k] = base[p * NFEAT + 1];
        pz[k] = base[p * NFEAT + 2];
        dist[k] = 3.402823466e38f;
    }

    __shared__ float s_val[32];
    __shared__ int   s_idx[32];
    __shared__ int   s_last;
    __shared__ float s_lx, s_ly, s_lz;

    int last = 0;
    if (tid == 0) { s_lx = base[0]; s_ly = base[1]; s_lz = base[2]; }
    __syncthreads();

    for (int i = 0; i < NPOINT; ++i) {
        if (tid == 0) fpsIdx[b * NPOINT + i] = last;
        float lx = s_lx, ly = s_ly, lz = s_lz;

        float bv = -1.0f; int bi = 0;
#pragma unroll
        for (int k = 0; k < 8; ++k) {
            float dx = px[k] - lx, dy = py[k] - ly, dz = pz[k] - lz;
            float nd = dx * dx + dy * dy + dz * dz;
            dist[k] = fminf(dist[k], nd);
            int p = tid + k * 1024;
            if (dist[k] > bv || (dist[k] == bv && p < bi)) { bv = dist[k]; bi = p; }
        }
        // wave32 argmax (first-max tie-break: lower index wins on equality)
        for (int off = 16; off > 0; off >>= 1) {
            float ov = __shfl_xor(bv, off, 32);
            int   oi = __shfl_xor(bi, off, 32);
            if (ov > bv || (ov == bv && oi < bi)) { bv = ov; bi = oi; }
        }
        if ((tid & 31) == 0) { s_val[tid >> 5] = bv; s_idx[tid >> 5] = bi; }
        __syncthreads();
        if (tid < 32) {
            bv = s_val[tid]; bi = s_idx[tid];
            for (int off = 16; off > 0; off >>= 1) {
                float ov = __shfl_xor(bv, off, 32);
                int   oi = __shfl_xor(bi, off, 32);
                if (ov > bv || (ov == bv && oi < bi)) { bv = ov; bi = oi; }
            }
            if (tid == 0) {
                s_last = bi;
                s_lx = base[bi * NFEAT + 0];
                s_ly = base[bi * NFEAT + 1];
                s_lz = base[bi * NFEAT + 2];
            }
        }
        __syncthreads();
        last = s_last;
    }
}

// ---------------------------------------------------------------------------
// 2a) SoA point arrays + |p|^2  (coalesced B-operand feeds for WMMA)
// ---------------------------------------------------------------------------
__global__ void prep_points_kernel(const float* __restrict__ in,
                                   float* __restrict__ px, float* __restrict__ py,
                                   float* __restrict__ pz, float* __restrict__ pn) {
    int b = blockIdx.y;
    int p = blockIdx.x * 256 + threadIdx.x;
    const float* r = in + ((size_t)b * NPTS + p) * NFEAT;
    float x = r[0], y = r[1], z = r[2];
    int o = b * NPTS + p;
    px[o] = x; py[o] = y; pz[o] = z; pn[o] = x * x + y * y + z * z;
}

// 2b) Gather centroids, emit new_xyz, SoA query arrays + |q|^2
__global__ void prep_queries_kernel(const float* __restrict__ in,
                                    const int* __restrict__ fpsIdx,
                                    float* __restrict__ qx, float* __restrict__ qy,
                                    float* __restrict__ qz, float* __restrict__ qn,
                                    float* __restrict__ outNewXyz) {
    int b = blockIdx.y;
    int q = blockIdx.x * 256 + threadIdx.x;
    int j = fpsIdx[b * NPOINT + q];
    const float* r = in + ((size_t)b * NPTS + j) * NFEAT;
    float x = r[0], y = r[1], z = r[2];
    int o = b * NPOINT + q;
    qx[o] = x; qy[o] = y; qz[o] = z; qn[o] = x * x + y * y + z * z;
    float* w = outNewXyz + (size_t)o * 3;
    w[0] = x; w[1] = y; w[2] = z;
}

// ---------------------------------------------------------------------------
// 3) Ball query via V_WMMA_F32_16X16X4_F32, two 16-col tiles per trip.
//    A(16x4) rows  = (-2qx, -2qy, -2qz, |q|^2)   [lane<16: K0/K1; lane>=16: K2/K3]
//    B(4x16) cols  = ( px,   py,   pz,  1   )
//    C init        = |p|^2 broadcast per column
//    => D[m][n] = |q_m|^2 + |p_n|^2 - 2 q_m.p_n  (exact d^2 tile)
//    ballot(D[r] < r2) gives rows r (low 16) and r+8 (high 16); lane L owns
//    row L and appends ascending column indices until 32 found.
// ---------------------------------------------------------------------------
__device__ __forceinline__ unsigned row_mask(const unsigned bal[8], int lane) {
    unsigned s = (unsigned)(lane & 7);
    unsigned br = 0;
#pragma unroll
    for (unsigned r = 0; r < 8; ++r) br |= (s == r) ? bal[r] : 0u;   // branchless
    unsigned shift = (unsigned)(lane & 8) << 1;                      // 0 or 16
    return (br >> shift) & 0xFFFFu;
}

__global__ __launch_bounds__(256) void ball_query_kernel(
    const float* __restrict__ px, const float* __restrict__ py,
    const float* __restrict__ pz, const float* __restrict__ pn,
    const float* __restrict__ qx, const float* __restrict__ qy,
    const float* __restrict__ qz, const float* __restrict__ qn,
    int* __restrict__ nbr) {
    const int b    = blockIdx.y;
    const int wave = threadIdx.x >> 5;
    const int lane = threadIdx.x & 31;
    const int q0   = (blockIdx.x * 8 + wave) * 16;
    const int m    = lane & 15;
    const bool hi  = lane >= 16;

    const float* PX = px + b * NPTS;   const float* PY = py + b * NPTS;
    const float* PZ = pz + b * NPTS;   const float* PN = pn + b * NPTS;
    const float* QX = qx + b * NPOINT; const float* QY = qy + b * NPOINT;
    const float* QZ = qz + b * NPOINT; const float* QN = qn + b * NPOINT;

    v2f A;
    A.x = hi ? (-2.0f * QZ[q0 + m]) : (-2.0f * QX[q0 + m]);
    A.y = hi ? (QN[q0 + m])         : (-2.0f * QY[q0 + m]);

    int cnt = 0;
    int first = -1;
    int* rowOut = nbr + (size_t)(b * NPOINT + q0 + lane) * NSAMPLE; // used when lane<16

    for (int j0 = 0; j0 < NPTS; j0 += 32) {
        __builtin_prefetch(PX + j0 + 32, 0, 0);   // global_prefetch_b8

        // ---- tile 0: columns j0 .. j0+15 ----
        v2f B0;
        B0.x = hi ? PZ[j0 + m] : PX[j0 + m];
        B0.y = hi ? 1.0f       : PY[j0 + m];
        float pnv0 = PN[j0 + m];
        v8f C0 = {pnv0, pnv0, pnv0, pnv0, pnv0, pnv0, pnv0, pnv0};

        // ---- tile 1: columns j0+16 .. j0+31 ----
        v2f B1;
        B1.x = hi ? PZ[j0 + 16 + m] : PX[j0 + 16 + m];
        B1.y = hi ? 1.0f            : PY[j0 + 16 + m];
        float pnv1 = PN[j0 + 16 + m];
        v8f C1 = {pnv1, pnv1, pnv1, pnv1, pnv1, pnv1, pnv1, pnv1};

        v8f D0 = __builtin_amdgcn_wmma_f32_16x16x4_f32(
            false, A, false, B0, (short)0, C0, false, false);
        v8f D1 = __builtin_amdgcn_wmma_f32_16x16x4_f32(
            false, A, false, B1, (short)0, C1, false, false);

        unsigned bal0[8], bal1[8];
#pragma unroll
        for (int r = 0; r < 8; ++r) bal0[r] = __builtin_amdgcn_ballot_w32(D0[r] < RADIUS2);
#pragma unroll
        for (int r = 0; r < 8; ++r) bal1[r] = __builtin_amdgcn_ballot_w32(D1[r] < RADIUS2);

        unsigned mask0 = row_mask(bal0, lane);
        unsigned mask1 = row_mask(bal1, lane);

        if (lane < 16) {
            while (mask0 && cnt < NSAMPLE) {
                int bit = __builtin_ctz(mask0);
                mask0 &= mask0 - 1u;
                int j = j0 + bit;
                if (cnt == 0) first = j;
                rowOut[cnt++] = j;
            }
            while (mask1 && cnt < NSAMPLE) {
                int bit = __builtin_ctz(mask1);
                mask1 &= mask1 - 1u;
                int j = j0 + 16 + bit;
                if (cnt == 0) first = j;
                rowOut[cnt++] = j;
            }
        }
        // wave-uniform early exit once all 16 rows are full
        if (__builtin_amdgcn_ballot_w32(lane < 16 && cnt < NSAMPLE) == 0u) break;
    }

    if (lane < 16) {
        if (first < 0) first = 0;
        while (cnt < NSAMPLE) rowOut[cnt++] = first;
    }
}

// ---------------------------------------------------------------------------
// 4) Gather + concat: new_points, idx (as float), grouped_xyz.
// ---------------------------------------------------------------------------
__global__ __launch_bounds__(256) void group_kernel(
    const float* __restrict__ in, const int* __restrict__ nbr,
    const float* __restrict__ qx, const float* __restrict__ qy,
    const float* __restrict__ qz,
    float* __restrict__ outNewPoints, float* __restrict__ outIdx,
    float* __restrict__ outGrpXyz) {
    size_t t = (size_t)blockIdx.x * 256 + threadIdx.x;   // b*NPOINT*NSAMPLE range
    int q = (int)((t / NSAMPLE) % NPOINT);
    int b = (int)(t / ((size_t)NSAMPLE * NPOINT));
    int j = nbr[t];
    const float* r = in + ((size_t)b * NPTS + j) * NFEAT;
    int o = b * NPOINT + q;
    float gx = r[0] - qx[o], gy = r[1] - qy[o], gz = r[2] - qz[o];

    float* np = outNewPoints + t * 16;
    np[0] = gx; np[1] = gy; np[2] = gz;
#pragma unroll
    for (int f = 3; f < 16; ++f) np[f] = r[f];

    float* g = outGrpXyz + t * 3;
    g[0] = gx; g[1] = gy; g[2] = gz;

    outIdx[t] = (float)j;
}

// ---------------------------------------------------------------------------
extern "C" void kernel_launch(void* const* d_in, const int* in_sizes, int n_in,
                              void* d_out, int out_size, void* d_ws, size_t ws_size,
                              hipStream_t stream) {
    const float* in = (const float*)d_in[0];

    char* ws = (char*)d_ws;
    size_t off = 0;
    auto alloc = [&](size_t bytes) { void* p = ws + off; off = (off + bytes + 255) & ~(size_t)255; return p; };

    int*   fpsIdx = (int*)  alloc((size_t)BATCH * NPOINT * sizeof(int));
    float* qx = (float*)alloc((size_t)BATCH * NPOINT * sizeof(float));
    float* qy = (float*)alloc((size_t)BATCH * NPOINT * sizeof(float));
    float* qz = (float*)alloc((size_t)BATCH * NPOINT * sizeof(float));
    float* qn = (float*)alloc((size_t)BATCH * NPOINT * sizeof(float));
    float* px = (float*)alloc((size_t)BATCH * NPTS * sizeof(float));
    float* py = (float*)alloc((size_t)BATCH * NPTS * sizeof(float));
    float* pz = (float*)alloc((size_t)BATCH * NPTS * sizeof(float));
    float* pn = (float*)alloc((size_t)BATCH * NPTS * sizeof(float));
    int*   nbr = (int*) alloc((size_t)BATCH * NPOINT * NSAMPLE * sizeof(int));

    float* out          = (float*)d_out;
    float* outNewXyz    = out;                                   // 4*2048*3
    float* outNewPoints = out + (size_t)BATCH * NPOINT * 3;      // 4*2048*32*16
    float* outIdx       = outNewPoints + (size_t)BATCH * NPOINT * NSAMPLE * 16;
    float* outGrpXyz    = outIdx + (size_t)BATCH * NPOINT * NSAMPLE;

    // 1) FPS: one block per batch
    fps_kernel<<<dim3(BATCH), dim3(1024), 0, stream>>>(in, fpsIdx);

    // 2) SoA prep (points independent of FPS; queries depend on it — same stream)
    prep_points_kernel<<<dim3(NPTS / 256, BATCH), dim3(256), 0, stream>>>(in, px, py, pz, pn);
    prep_queries_kernel<<<dim3(NPOINT / 256, BATCH), dim3(256), 0, stream>>>(
        in, fpsIdx, qx, qy, qz, qn, outNewXyz);

    // 3) Ball query (WMMA): 8 waves/block, 16 queries/wave, 32 cols/trip
    ball_query_kernel<<<dim3(NPOINT / (8 * 16), BATCH), dim3(256), 0, stream>>>(
        px, py, pz, pn, qx, qy, qz, qn, nbr);

    // 4) Gather/concat outputs
    group_kernel<<<dim3((BATCH * NPOINT * NSAMPLE) / 256), dim3(256), 0, stream>>>(
        in, nbr, qx, qy, qz, outNewPoints, outIdx, outGrpXyz);
}